// SSND_42331197669501
// MI455X (gfx1250) — compile-verified
//
#include <hip/hip_runtime.h>
#include <hip/hip_bf16.h>
#include <math.h>

// Problem constants (match reference)
constexpr int Bb   = 2;
constexpr int Hh   = 64;
constexpr int Wd   = 64;
constexpr int Ll   = Hh * Wd;        // 4096
constexpr int DM   = 192;            // d_model
constexpr int DI   = 384;            // d_inner
constexpr int NST  = 16;             // d_state
constexpr int RK   = 12;             // dt_rank
constexpr int KD   = 4;              // directions
constexpr int CPROJ = RK + 2 * NST;  // 44
constexpr int NC   = 64;             // scan chunks
constexpr int CL   = Ll / NC;        // 64 steps per chunk

typedef __attribute__((ext_vector_type(2))) float v2f;
typedef __attribute__((ext_vector_type(8))) float v8f;

#if __has_builtin(__builtin_amdgcn_global_load_async_to_lds_b32)
#define USE_ASYNC_LDS 1
#else
#define USE_ASYNC_LDS 0
#endif

#if USE_ASYNC_LDS
typedef __attribute__((address_space(1))) int gint_t;
typedef __attribute__((address_space(3))) int lint_t;
__device__ __forceinline__ void async_copy_f32(const float* g, float* l) {
    __builtin_amdgcn_global_load_async_to_lds_b32(
        (gint_t*)(g), (lint_t*)(l), 0, 0);
}
#if __has_builtin(__builtin_amdgcn_s_wait_asynccnt)
#define WAIT_ASYNC(n) __builtin_amdgcn_s_wait_asynccnt(n)
#else
#define WAIT_ASYNC(n) asm volatile("s_wait_asynccnt %0" :: "i"(n))
#endif
#endif

// scan index -> row-major spatial position for direction k
__device__ __forceinline__ int scan_pos(int k, int l) {
    switch (k & 3) {
        case 0: return l;                                            // row-major
        case 1: { int w = l >> 6, h = l & 63; return h * Wd + w; }   // col-major (l = w*H + h)
        case 2: return Ll - 1 - l;                                   // reversed row-major
        default: { int lr = Ll - 1 - l; int w = lr >> 6, h = lr & 63; return h * Wd + w; }
    }
}

// ---------------------------------------------------------------------------
// fp32 WMMA GEMM, 16Mx64N strip per wave (4 accumulators, A-frag reused 4x):
//   C[m,n] = sum_k A[m*lda+k] * W[n*ldw+k]      (N must be a multiple of 64)
// ---------------------------------------------------------------------------
__global__ void wmma_gemm_nt4(const float* __restrict__ A, const float* __restrict__ W,
                              float* __restrict__ C, int M, int stripsN, int K,
                              int lda, int ldw, int ldc) {
    int wave = (int)((blockIdx.x * blockDim.x + threadIdx.x) >> 5);
    int lane = threadIdx.x & 31;
    int tm = wave / stripsN, sn = wave % stripsN;
    int m0 = tm * 16, n0 = sn * 64;
    if (m0 >= M) return;

    int khalf = (lane >> 4) ? 2 : 0;   // lanes 16-31 carry K=2,3
    const float* arow = A + (size_t)(m0 + (lane & 15)) * lda + khalf;
    const float* wrow = W + (size_t)(n0 + (lane & 15)) * ldw + khalf;

    v8f acc0 = {}, acc1 = {}, acc2 = {}, acc3 = {};
    for (int k0 = 0; k0 < K; k0 += 4) {
        v2f a, b0, b1, b2, b3;
        a.x  = arow[k0];              a.y  = arow[k0 + 1];
        b0.x = wrow[k0];              b0.y = wrow[k0 + 1];
        b1.x = wrow[16 * ldw + k0];   b1.y = wrow[16 * ldw + k0 + 1];
        b2.x = wrow[32 * ldw + k0];   b2.y = wrow[32 * ldw + k0 + 1];
        b3.x = wrow[48 * ldw + k0];   b3.y = wrow[48 * ldw + k0 + 1];
        acc0 = __builtin_amdgcn_wmma_f32_16x16x4_f32(false, a, false, b0, (short)0, acc0, false, false);
        acc1 = __builtin_amdgcn_wmma_f32_16x16x4_f32(false, a, false, b1, (short)0, acc1, false, false);
        acc2 = __builtin_amdgcn_wmma_f32_16x16x4_f32(false, a, false, b2, (short)0, acc2, false, false);
        acc3 = __builtin_amdgcn_wmma_f32_16x16x4_f32(false, a, false, b3, (short)0, acc3, false, false);
    }
    int mbase = m0 + ((lane >> 4) << 3);
    int ncol  = n0 + (lane & 15);
    #pragma unroll
    for (int i = 0; i < 8; ++i) {
        float* crow = C + (size_t)(mbase + i) * ldc + ncol;
        crow[0]  = acc0[i];
        crow[16] = acc1[i];
        crow[32] = acc2[i];
        crow[48] = acc3[i];
    }
}

// ---------------------------------------------------------------------------
// Depthwise 3x3 conv (SAME) + bias + SiLU.
// Input: xz rows (b*L + l) x 768, channels 0..383 are xi.
// Output: xc laid out (B, DI, L) channel-major for the scan.
// ---------------------------------------------------------------------------
__global__ void dwconv_silu(const float* __restrict__ xz, const float* __restrict__ cw,
                            const float* __restrict__ cb, float* __restrict__ xc) {
    int idx = blockIdx.x * blockDim.x + threadIdx.x;
    if (idx >= Bb * DI * Ll) return;
    int l = idx % Ll; int t = idx / Ll; int d = t % DI; int b = t / DI;
    int h = l >> 6, w = l & 63;
    float s = cb[d];
    #pragma unroll
    for (int dy = -1; dy <= 1; ++dy) {
        #pragma unroll
        for (int dx = -1; dx <= 1; ++dx) {
            int hh = h + dy, ww = w + dx;
            if (hh >= 0 && hh < Hh && ww >= 0 && ww < Wd) {
                float v = xz[(size_t)(b * Ll + hh * Wd + ww) * (2 * DI) + d];
                s += v * cw[d * 9 + (dy + 1) * 3 + (dx + 1)];
            }
        }
    }
    xc[idx] = s * (1.f / (1.f + __expf(-s)));   // SiLU
}

// ---------------------------------------------------------------------------
// x_proj GEMM with gathered A: x_dbl[b,k,l,c] = sum_d xpw[k,c,d]*xc[b,d,pos_k(l)]
// One wave per 16-row tile, all 3 N-tiles (gathered A-frag reused 3x).
// Tail tile (cols 32..43 of 44) handled branch-free via clamped row + mask.
// ---------------------------------------------------------------------------
__global__ void xproj_gemm(const float* __restrict__ xc, const float* __restrict__ xpw,
                           float* __restrict__ xdbl) {
    int bk = blockIdx.y; int b = bk >> 2, k = bk & 3;
    int wave = (int)((blockIdx.x * blockDim.x + threadIdx.x) >> 5);  // m-tile 0..255
    int lane = threadIdx.x & 31;
    int m0 = wave * 16;
    int mrow = m0 + (lane & 15);
    int pos  = scan_pos(k, mrow);
    int khalf = (lane >> 4) ? 2 : 0;

    const float* Wk  = xpw + (size_t)k * CPROJ * DI;
    const float* xcb = xc + (size_t)b * DI * Ll + pos;

    int r0 = lane & 15;
    int r2 = 32 + r0;
    float msk2 = (r2 < CPROJ) ? 1.f : 0.f;
    int r2c = (r2 < CPROJ) ? r2 : (CPROJ - 1);
    const float* w0 = Wk + (size_t)r0 * DI + khalf;
    const float* w1 = Wk + (size_t)(16 + r0) * DI + khalf;
    const float* w2 = Wk + (size_t)r2c * DI + khalf;

    v8f acc0 = {}, acc1 = {}, acc2 = {};
    for (int k0 = 0; k0 < DI; k0 += 4) {
        v2f a, b0, b1, b2;
        a.x  = xcb[(size_t)(k0 + khalf) * Ll];
        a.y  = xcb[(size_t)(k0 + khalf + 1) * Ll];
        b0.x = w0[k0];        b0.y = w0[k0 + 1];
        b1.x = w1[k0];        b1.y = w1[k0 + 1];
        b2.x = w2[k0] * msk2; b2.y = w2[k0 + 1] * msk2;
        acc0 = __builtin_amdgcn_wmma_f32_16x16x4_f32(false, a, false, b0, (short)0, acc0, false, false);
        acc1 = __builtin_amdgcn_wmma_f32_16x16x4_f32(false, a, false, b1, (short)0, acc1, false, false);
        acc2 = __builtin_amdgcn_wmma_f32_16x16x4_f32(false, a, false, b2, (short)0, acc2, false, false);
    }
    int mbase = m0 + ((lane >> 4) << 3);
    int ncol  = lane & 15;
    float* Cout = xdbl + (size_t)bk * Ll * CPROJ;
    #pragma unroll
    for (int i = 0; i < 8; ++i) {
        float* crow = Cout + (size_t)(mbase + i) * CPROJ + ncol;
        crow[0]  = acc0[i];
        crow[16] = acc1[i];
        if (ncol < CPROJ - 32) crow[32] = acc2[i];
    }
}

// ---------------------------------------------------------------------------
// Chunked selective scan. h_l = a_l*h_{l-1} + b_l is associative, so:
//   pass A (PASSB=false): per chunk from h=0, record (prod a, h_end) per (d,n)
//   combine: sequential carry over 64 chunks for 49k independent (b,k,d,n)
//   pass B (PASSB=true): replay chunk from corrected h_init, emit y
// One block of 384 threads per (b,k,chunk). The shared 44-wide (dtr|B|C) row
// is double-buffered through LDS via async global->LDS copies when available.
// ---------------------------------------------------------------------------
template <bool PASSB>
__global__ void scan_chunk(const float* __restrict__ xc, const float* __restrict__ xdbl,
                           const float* __restrict__ dtw, const float* __restrict__ dtb,
                           const float* __restrict__ A_logs, const float* __restrict__ Ds,
                           float* __restrict__ carry, float* __restrict__ y4) {
    int blk = blockIdx.x;
    int bk = blk / NC, c = blk % NC;
    int b = bk >> 2, k = bk & 3;
    int d = threadIdx.x;                  // 0..383
    __shared__ float srow[2][CPROJ];      // [0..11] dtr, [12..27] B, [28..43] C

    float Arow[NST];
    #pragma unroll
    for (int n = 0; n < NST; ++n)
        Arow[n] = -__expf(A_logs[(size_t)(k * DI + d) * NST + n]);
    float wdt[RK];
    #pragma unroll
    for (int r = 0; r < RK; ++r)
        wdt[r] = dtw[(size_t)(k * DI + d) * RK + r];
    float bias = dtb[k * DI + d];
    float Dsk  = PASSB ? Ds[k * DI + d] : 0.f;

    float* cr = carry + ((size_t)(bk * DI + d) * NC + c) * (2 * NST);

    float h[NST], P[NST];
    #pragma unroll
    for (int n = 0; n < NST; ++n) {
        h[n] = PASSB ? cr[n] : 0.f;       // combine stored h_init in P-slot
        P[n] = 1.f;
    }

    const float* xcb = xc + (size_t)(b * DI + d) * Ll;
    const float* row = xdbl + ((size_t)bk * Ll + (size_t)c * CL) * CPROJ;
    float* yout = y4 + (size_t)b * Ll * DI * 4 + (size_t)d * 4 + k;

#if USE_ASYNC_LDS
    if (d < CPROJ) async_copy_f32(row + d, &srow[0][d]);
#endif
    for (int t = 0; t < CL; ++t) {
        int l = c * CL + t;
#if USE_ASYNC_LDS
        int tn = (t + 1 < CL) ? t + 1 : t;     // harmless re-fetch on last iter
        if (d < CPROJ) async_copy_f32(row + (size_t)tn * CPROJ + d, &srow[(t + 1) & 1][d]);
        WAIT_ASYNC(1);                          // row t landed; t+1 may be in flight
        __syncthreads();
#else
        if (d < CPROJ) srow[t & 1][d] = row[(size_t)t * CPROJ + d];
        __syncthreads();
#endif
        const float* sr = srow[t & 1];

        float dot = bias;
        #pragma unroll
        for (int r = 0; r < RK; ++r) dot += wdt[r] * sr[r];
        float delta = (dot > 20.f) ? dot : __logf(1.f + __expf(dot));  // softplus

        int pos = scan_pos(k, l);
        float u  = xcb[pos];
        float du = delta * u;
        float y  = Dsk * u;
        #pragma unroll
        for (int n = 0; n < NST; ++n) {
            float a = __expf(delta * Arow[n]);
            h[n] = a * h[n] + du * sr[RK + n];
            if (PASSB) y += h[n] * sr[RK + NST + n];
            else       P[n] *= a;
        }
        __syncthreads();
        if (PASSB) yout[(size_t)pos * DI * 4] = y;
    }
    if (!PASSB) {
        #pragma unroll
        for (int n = 0; n < NST; ++n) { cr[n] = P[n]; cr[NST + n] = h[n]; }
    }
}

// Sequential inter-chunk carry: h_init[c] written over the P-slot (in place).
__global__ void scan_combine(float* __restrict__ carry) {
    int idx = blockIdx.x * blockDim.x + threadIdx.x;   // (bk*DI+d)*NST + n
    if (idx >= Bb * KD * DI * NST) return;
    int n = idx & (NST - 1);
    size_t bkd = (size_t)idx / NST;
    float* cr = carry + bkd * NC * (2 * NST);
    float h = 0.f;
    for (int c = 0; c < NC; ++c) {
        float Pv = cr[c * 2 * NST + n];
        float Ev = cr[c * 2 * NST + NST + n];
        cr[c * 2 * NST + n] = h;          // h_init for chunk c
        h = Pv * h + Ev;
    }
}

// ---------------------------------------------------------------------------
// Merge 4 directions + LayerNorm(DI) + SiLU(z) gate. One wave per token.
// ---------------------------------------------------------------------------
__global__ void ln_gate(const float* __restrict__ y4, const float* __restrict__ xz,
                        const float* __restrict__ gamma, const float* __restrict__ beta,
                        float* __restrict__ yg) {
    int wave = (int)((blockIdx.x * blockDim.x + threadIdx.x) >> 5);
    int lane = threadIdx.x & 31;
    if (wave >= Bb * Ll) return;

    const float* yrow = y4 + (size_t)wave * DI * 4;
    float vals[DI / 32];
    float sum = 0.f;
    #pragma unroll
    for (int i = 0; i < DI / 32; ++i) {
        int c = lane + i * 32;
        const float* p = yrow + (size_t)c * 4;
        float v = p[0] + p[1] + p[2] + p[3];
        vals[i] = v; sum += v;
    }
    #pragma unroll
    for (int off = 16; off > 0; off >>= 1) sum += __shfl_xor(sum, off, 32);
    float mu = sum * (1.f / DI);
    float var = 0.f;
    #pragma unroll
    for (int i = 0; i < DI / 32; ++i) { float t = vals[i] - mu; var += t * t; }
    #pragma unroll
    for (int off = 16; off > 0; off >>= 1) var += __shfl_xor(var, off, 32);
    float rstd = rsqrtf(var * (1.f / DI) + 1e-5f);

    const float* zrow = xz + (size_t)wave * (2 * DI) + DI;
    float* o = yg + (size_t)wave * DI;
    #pragma unroll
    for (int i = 0; i < DI / 32; ++i) {
        int c = lane + i * 32;
        float zn = zrow[c];
        float g = zn * (1.f / (1.f + __expf(-zn)));             // SiLU gate
        o[c] = ((vals[i] - mu) * rstd * gamma[c] + beta[c]) * g;
    }
}

// ---------------------------------------------------------------------------
extern "C" void kernel_launch(void* const* d_in, const int* in_sizes, int n_in,
                              void* d_out, int out_size, void* d_ws, size_t ws_size,
                              hipStream_t stream) {
    const float* x         = (const float*)d_in[0];   // (B,H,W,192)
    const float* in_proj_w = (const float*)d_in[1];   // (768,192)
    const float* conv_w    = (const float*)d_in[2];   // (384,1,3,3)
    const float* conv_b    = (const float*)d_in[3];   // (384)
    const float* x_proj_w  = (const float*)d_in[4];   // (4,44,384)
    const float* dt_w      = (const float*)d_in[5];   // (4,384,12)
    const float* dt_b      = (const float*)d_in[6];   // (4,384)
    const float* A_logs    = (const float*)d_in[7];   // (4,384,16)
    const float* Ds        = (const float*)d_in[8];   // (4,384)
    const float* gamma     = (const float*)d_in[9];   // (384)
    const float* beta      = (const float*)d_in[10];  // (384)
    const float* out_w     = (const float*)d_in[11];  // (192,384)
    float* out = (float*)d_out;
    float* ws  = (float*)d_ws;

    // workspace layout (floats)
    float* xz    = ws;                 // 8192 x 768            = 6,291,456
    float* xc    = ws + 6291456;       // 2 x 384 x 4096        = 3,145,728
    float* xdbl  = ws + 9437184;       // 8 x 4096 x 44         = 1,441,792
    float* y4    = ws + 10878976;      // 8192 x 384 x 4        = 12,582,912
    float* yg    = ws + 23461888;      // 8192 x 384            = 3,145,728
    float* carry = ws + 26607616;      // 8 x 384 x 64 x 32     = 6,291,456

    const int M = Bb * Ll;             // 8192 tokens

    // 1) in_proj: (8192,192) x (768,192)^T -> (8192,768); 512*12 strips = 6144 waves
    wmma_gemm_nt4<<<768, 256, 0, stream>>>(x, in_proj_w, xz, M, 12, DM, DM, DM, 2 * DI);

    // 2) depthwise conv + SiLU -> xc (B,DI,L)
    dwconv_silu<<<(Bb * DI * Ll) / 256, 256, 0, stream>>>(xz, conv_w, conv_b, xc);

    // 3) x_proj per (b,k): (4096,384-gathered) x (44,384)^T -> (4096,44)
    dim3 gx(32, Bb * KD);              // 256 m-tile waves per (b,k)
    xproj_gemm<<<gx, 256, 0, stream>>>(xc, x_proj_w, xdbl);

    // 4) chunked selective scan: 512 blocks x 384 threads, then carry, then replay
    scan_chunk<false><<<Bb * KD * NC, DI, 0, stream>>>(xc, xdbl, dt_w, dt_b, A_logs, Ds, carry, y4);
    scan_combine<<<(Bb * KD * DI * NST) / 256, 256, 0, stream>>>(carry);
    scan_chunk<true><<<Bb * KD * NC, DI, 0, stream>>>(xc, xdbl, dt_w, dt_b, A_logs, Ds, carry, y4);

    // 5) merge + layernorm + gate: one wave per token
    ln_gate<<<(M * 32) / 256, 256, 0, stream>>>(y4, xz, gamma, beta, yg);

    // 6) out_proj: (8192,384) x (192,384)^T -> (8192,192); 512*3 strips = 1536 waves
    wmma_gemm_nt4<<<192, 256, 0, stream>>>(yg, out_w, out, M, 3, DI, DI, DI, DM);
}